// DSSMamba_6588479832214
// MI455X (gfx1250) — compile-verified
//
#include <hip/hip_runtime.h>
#include <hip/hip_bf16.h>

// ---- problem constants (f32 reference) ----
#define BSZ      4
#define LSEQ     900
#define DMODEL   256
#define DSTATE   16
#define DINNER   512
#define DTRANK   16
#define ROWS     (BSZ * LSEQ)          // 3600 (= 225 * 16, exact M tiles)

typedef __attribute__((ext_vector_type(2))) float v2f;
typedef __attribute__((ext_vector_type(8))) float v8f;

// f32 WMMA: D(16x16,f32) = A(16x4,f32) * B(4x16,f32) + C
__device__ __forceinline__ v8f wmma4(v2f a, v2f b, v8f c) {
  return __builtin_amdgcn_wmma_f32_16x16x4_f32(false, a, false, b, (short)0, c,
                                               false, false);
}

__device__ __forceinline__ float sigmoid_(float x) { return 1.0f / (1.0f + __expf(-x)); }
__device__ __forceinline__ float silu_(float x)    { return x * sigmoid_(x); }

// ---------------------------------------------------------------------------
// Kernel 1: xz = hidden @ W_in^T, then SiLU both halves.
// One wave (32 threads) per (mtile, ngroup-of-64). Grid = 225*16 blocks.
// ---------------------------------------------------------------------------
__global__ __launch_bounds__(32)
void k_inproj(const float* __restrict__ hid, const float* __restrict__ Win,
              float* __restrict__ xbuf, float* __restrict__ zgbuf) {
  const int lane = threadIdx.x & 31;
  const int half = lane >> 4, l15 = lane & 15;
  const int mtile = blockIdx.x >> 4;    // 0..224
  const int ngrp  = blockIdx.x & 15;    // 0..15 (64 cols each)
  const int m = mtile * 16 + l15;
  const float* __restrict__ arow = hid + (size_t)m * DMODEL;

  v8f acc[4] = {};
  for (int k0 = 0; k0 < DMODEL; k0 += 4) {
    const int ka = k0 + 2 * half;
    __builtin_prefetch(arow + k0 + 64, 0, 0);   // global_prefetch_b8
    v2f a; a.x = arow[ka]; a.y = arow[ka + 1];
#pragma unroll
    for (int j = 0; j < 4; ++j) {
      const int n = ngrp * 64 + j * 16 + l15;
      const float* __restrict__ brow = Win + (size_t)n * DMODEL;
      v2f b; b.x = brow[ka]; b.y = brow[ka + 1];
      acc[j] = wmma4(a, b, acc[j]);
    }
  }
#pragma unroll
  for (int j = 0; j < 4; ++j) {
    const int n = ngrp * 64 + j * 16 + l15;
#pragma unroll
    for (int r = 0; r < 8; ++r) {
      const int row = mtile * 16 + r + 8 * half;
      const float s = silu_(acc[j][r]);
      if (n < DINNER) xbuf [(size_t)row * DINNER + n]            = s;
      else            zgbuf[(size_t)row * DINNER + (n - DINNER)] = s;
    }
  }
}

// ---------------------------------------------------------------------------
// Kernel 2: x_dbl = x @ W_xproj^T  (N = 48 = 3 tiles). Grid = 225*3 waves.
// ---------------------------------------------------------------------------
__global__ __launch_bounds__(32)
void k_xproj(const float* __restrict__ xbuf, const float* __restrict__ Wx,
             float* __restrict__ dtr, float* __restrict__ Bm,
             float* __restrict__ Cm) {
  const int lane = threadIdx.x & 31;
  const int half = lane >> 4, l15 = lane & 15;
  const int mtile = blockIdx.x / 3, ntile = blockIdx.x % 3;
  const int m = mtile * 16 + l15;
  const float* __restrict__ arow = xbuf + (size_t)m * DINNER;
  const int n = ntile * 16 + l15;
  const float* __restrict__ brow = Wx + (size_t)n * DINNER;

  v8f acc = {};
  for (int k0 = 0; k0 < DINNER; k0 += 4) {
    const int ka = k0 + 2 * half;
    __builtin_prefetch(arow + k0 + 64, 0, 0);
    v2f a; a.x = arow[ka]; a.y = arow[ka + 1];
    v2f b; b.x = brow[ka]; b.y = brow[ka + 1];
    acc = wmma4(a, b, acc);
  }
  float* __restrict__ outp = (ntile == 0) ? dtr : ((ntile == 1) ? Bm : Cm);
#pragma unroll
  for (int r = 0; r < 8; ++r) {
    const int row = mtile * 16 + r + 8 * half;
    outp[(size_t)row * 16 + l15] = acc[r];
  }
}

// ---------------------------------------------------------------------------
// Kernel 3: delta = softplus(dt_r @ W_dt^T + b_dt). K = 16 (4 wmma steps).
// Grid = 225*8 waves (8 ngroups of 64 over DINNER=512).
// ---------------------------------------------------------------------------
__global__ __launch_bounds__(32)
void k_delta(const float* __restrict__ dtr, const float* __restrict__ Wdt,
             const float* __restrict__ bdt, float* __restrict__ delta) {
  const int lane = threadIdx.x & 31;
  const int half = lane >> 4, l15 = lane & 15;
  const int mtile = blockIdx.x >> 3;
  const int ngrp  = blockIdx.x & 7;
  const int m = mtile * 16 + l15;
  const float* __restrict__ arow = dtr + (size_t)m * DTRANK;

  v8f acc[4] = {};
  for (int k0 = 0; k0 < DTRANK; k0 += 4) {
    const int ka = k0 + 2 * half;
    v2f a; a.x = arow[ka]; a.y = arow[ka + 1];
#pragma unroll
    for (int j = 0; j < 4; ++j) {
      const int n = ngrp * 64 + j * 16 + l15;
      const float* __restrict__ brow = Wdt + (size_t)n * DTRANK;
      v2f b; b.x = brow[ka]; b.y = brow[ka + 1];
      acc[j] = wmma4(a, b, acc[j]);
    }
  }
#pragma unroll
  for (int j = 0; j < 4; ++j) {
    const int n = ngrp * 64 + j * 16 + l15;
    const float bias = bdt[n];
#pragma unroll
    for (int r = 0; r < 8; ++r) {
      const int row = mtile * 16 + r + 8 * half;
      const float v = acc[j][r] + bias;
      delta[(size_t)row * DINNER + n] = (v > 20.0f) ? v : log1pf(__expf(v));
    }
  }
}

// ---------------------------------------------------------------------------
// Kernel 4: bidirectional selective scan. Grid = 2 dir * 4 batch * 4 chunks
// = 32 blocks x 128 threads; each thread owns one d-channel (16 states).
// Software-pipelined: step s prefetches step s+1's B/C rows into the other
// LDS buffer while computing from the current one; one barrier per step.
// Critical-path exps use v_exp_f32 (TRANS, co-executes with VALU).
// ---------------------------------------------------------------------------
__global__ __launch_bounds__(128)
void k_scan(const float* __restrict__ xbuf, const float* __restrict__ delta,
            const float* __restrict__ Bm, const float* __restrict__ Cm,
            const float* __restrict__ Alogf, const float* __restrict__ Alogb,
            const float* __restrict__ Df, const float* __restrict__ Db,
            float* __restrict__ yf, float* __restrict__ yb) {
  const int dir   = blockIdx.x >> 4;
  const int rem   = blockIdx.x & 15;
  const int batch = rem >> 2;
  const int d     = (rem & 3) * 128 + threadIdx.x;

  const float* __restrict__ Alog = dir ? Alogb : Alogf;
  const float* __restrict__ Dv   = dir ? Db : Df;
  float* __restrict__ yout       = dir ? yb : yf;

  float Arow[DSTATE];
#pragma unroll
  for (int n = 0; n < DSTATE; ++n) Arow[n] = -__expf(Alog[(size_t)d * DSTATE + n]);
  const float Dd = Dv[d];

  float h[DSTATE];
#pragma unroll
  for (int n = 0; n < DSTATE; ++n) h[n] = 0.0f;

  __shared__ float sB[2][DSTATE];
  __shared__ float sC[2][DSTATE];

  // prologue: stage step 0 into buffer 0
  {
    const int t0 = dir ? (LSEQ - 1) : 0;
    const size_t row0 = (size_t)batch * LSEQ + t0;
    if (threadIdx.x < 16)      sB[0][threadIdx.x]      = Bm[row0 * 16 + threadIdx.x];
    else if (threadIdx.x < 32) sC[0][threadIdx.x - 16] = Cm[row0 * 16 + threadIdx.x - 16];
  }
  __syncthreads();

  for (int s = 0; s < LSEQ; ++s) {
    const int t = dir ? (LSEQ - 1 - s) : s;
    const size_t row = (size_t)batch * LSEQ + t;
    const int p = s & 1;

    // prefetch next step's B/C into the other buffer (hidden behind compute)
    if (s + 1 < LSEQ) {
      const int tn = dir ? (LSEQ - 2 - s) : (s + 1);
      const size_t rown = (size_t)batch * LSEQ + tn;
      if (threadIdx.x < 16)      sB[p ^ 1][threadIdx.x]      = Bm[rown * 16 + threadIdx.x];
      else if (threadIdx.x < 32) sC[p ^ 1][threadIdx.x - 16] = Cm[rown * 16 + threadIdx.x - 16];
    }

    const float dt = delta[row * DINNER + d];
    const float u  = xbuf [row * DINNER + d];
    const float du = dt * u;
    float accy = 0.0f;
#pragma unroll
    for (int n = 0; n < DSTATE; ++n) {
      const float dA = __expf(dt * Arow[n]);     // v_exp_f32 (TRANS)
      h[n] = dA * h[n] + du * sB[p][n];
      accy += h[n] * sC[p][n];
    }
    yout[row * DINNER + d] = accy + Dd * u;
    __syncthreads();
  }
}

// ---------------------------------------------------------------------------
// Kernel 5: ycat = [yf * silu(z), yb * silu(z)]   (3600 x 1024)
// ---------------------------------------------------------------------------
__global__ __launch_bounds__(256)
void k_ycat(const float* __restrict__ yf, const float* __restrict__ yb,
            const float* __restrict__ zg, float* __restrict__ ycat) {
  const size_t idx = (size_t)blockIdx.x * 256 + threadIdx.x;
  const int row = (int)(idx >> 10);
  const int n   = (int)(idx & 1023);
  const float z = zg[(size_t)row * DINNER + (n & (DINNER - 1))];
  const float v = (n < DINNER) ? yf[(size_t)row * DINNER + n]
                               : yb[(size_t)row * DINNER + (n - DINNER)];
  ycat[idx] = v * z;
}

// ---------------------------------------------------------------------------
// Kernel 6: per-batch column mean over L. Grid = 16 blocks x 256.
// ---------------------------------------------------------------------------
__global__ __launch_bounds__(256)
void k_mean(const float* __restrict__ ycat, float* __restrict__ mean) {
  const int b = blockIdx.x >> 2;
  const int n = ((blockIdx.x & 3) << 8) + threadIdx.x;
  const float* __restrict__ base = ycat + (size_t)b * LSEQ * 1024 + n;
  float s = 0.0f;
  for (int l = 0; l < LSEQ; ++l) s += base[(size_t)l * 1024];
  mean[b * 1024 + n] = s * (1.0f / (float)LSEQ);
}

// ---------------------------------------------------------------------------
// Kernels 7/8: tiny gating GEMVs (4 x 1024) — VALU dot products.
// ---------------------------------------------------------------------------
__global__ __launch_bounds__(256)
void k_gate(const float* __restrict__ W, const float* __restrict__ bias,
            const float* __restrict__ vin, float* __restrict__ vout, int act) {
  const int b = blockIdx.x >> 2;
  const int j = ((blockIdx.x & 3) << 8) + threadIdx.x;
  const float* __restrict__ w = W + (size_t)j * 1024;
  const float* __restrict__ x = vin + (size_t)b * 1024;
  float s = bias[j];
  for (int k = 0; k < 1024; ++k) s += w[k] * x[k];
  vout[b * 1024 + j] = act ? sigmoid_(s) : s;
}

// ---------------------------------------------------------------------------
// Kernel 9: out = (ycat * g[batch]) @ W_out^T.  Grid = 225*4 waves.
// ---------------------------------------------------------------------------
__global__ __launch_bounds__(32)
void k_outproj(const float* __restrict__ ycat, const float* __restrict__ g,
               const float* __restrict__ Wout, float* __restrict__ out) {
  const int lane = threadIdx.x & 31;
  const int half = lane >> 4, l15 = lane & 15;
  const int mtile = blockIdx.x >> 2;
  const int ngrp  = blockIdx.x & 3;
  const int m  = mtile * 16 + l15;
  const int bm = m / LSEQ;
  const float* __restrict__ arow = ycat + (size_t)m * 1024;
  const float* __restrict__ grow = g + (size_t)bm * 1024;

  v8f acc[4] = {};
  for (int k0 = 0; k0 < 1024; k0 += 4) {
    const int ka = k0 + 2 * half;
    __builtin_prefetch(arow + k0 + 64, 0, 0);
    v2f a; a.x = arow[ka] * grow[ka]; a.y = arow[ka + 1] * grow[ka + 1];
#pragma unroll
    for (int j = 0; j < 4; ++j) {
      const int n = ngrp * 64 + j * 16 + l15;
      const float* __restrict__ brow = Wout + (size_t)n * 1024;
      v2f b; b.x = brow[ka]; b.y = brow[ka + 1];
      acc[j] = wmma4(a, b, acc[j]);
    }
  }
#pragma unroll
  for (int j = 0; j < 4; ++j) {
    const int n = ngrp * 64 + j * 16 + l15;
#pragma unroll
    for (int r = 0; r < 8; ++r) {
      const int row = mtile * 16 + r + 8 * half;
      out[(size_t)row * DMODEL + n] = acc[j][r];
    }
  }
}

// ---------------------------------------------------------------------------
extern "C" void kernel_launch(void* const* d_in, const int* in_sizes, int n_in,
                              void* d_out, int out_size, void* d_ws, size_t ws_size,
                              hipStream_t stream) {
  const float* hid  = (const float*)d_in[0];   // (4,900,256)
  const float* Win  = (const float*)d_in[1];   // (1024,256)
  const float* Wx   = (const float*)d_in[2];   // (48,512)
  const float* Wdt  = (const float*)d_in[3];   // (512,16)
  const float* bdt  = (const float*)d_in[4];   // (512)
  const float* Alf  = (const float*)d_in[5];   // (512,16)
  const float* Alb  = (const float*)d_in[6];   // (512,16)
  const float* Df   = (const float*)d_in[7];   // (512)
  const float* Db   = (const float*)d_in[8];   // (512)
  const float* Wg   = (const float*)d_in[9];   // (1024,1024)
  const float* bg   = (const float*)d_in[10];  // (1024)
  const float* Wgt  = (const float*)d_in[11];  // (1024,1024)
  const float* bgt  = (const float*)d_in[12];  // (1024)
  const float* Wout = (const float*)d_in[13];  // (256,1024)

  float* ws = (float*)d_ws;
  const size_t R = ROWS;
  float* xbuf  = ws;                    // R*512
  float* zgb   = xbuf  + R * 512;       // R*512
  float* dtr   = zgb   + R * 512;       // R*16
  float* Bm    = dtr   + R * 16;        // R*16
  float* Cm    = Bm    + R * 16;        // R*16
  float* delta = Cm    + R * 16;        // R*512
  float* yf    = delta + R * 512;       // R*512
  float* yb    = yf    + R * 512;       // R*512
  float* ycat  = yb    + R * 512;       // R*1024
  float* mean  = ycat  + R * 1024;      // 4096
  float* tmp   = mean  + 4096;          // 4096
  float* gv    = tmp   + 4096;          // 4096

  k_inproj <<<225 * 16, 32, 0, stream>>>(hid, Win, xbuf, zgb);
  k_xproj  <<<225 * 3,  32, 0, stream>>>(xbuf, Wx, dtr, Bm, Cm);
  k_delta  <<<225 * 8,  32, 0, stream>>>(dtr, Wdt, bdt, delta);
  k_scan   <<<32, 128, 0, stream>>>(xbuf, delta, Bm, Cm, Alf, Alb, Df, Db, yf, yb);
  k_ycat   <<<(ROWS * 1024) / 256, 256, 0, stream>>>(yf, yb, zgb, ycat);
  k_mean   <<<16, 256, 0, stream>>>(ycat, mean);
  k_gate   <<<16, 256, 0, stream>>>(Wg,  bg,  mean, tmp, 0);
  k_gate   <<<16, 256, 0, stream>>>(Wgt, bgt, tmp,  gv,  1);
  k_outproj<<<225 * 4, 32, 0, stream>>>(ycat, gv, Wout, (float*)d_out);
}